// PointNetSetAbstraction_85117661872237
// MI455X (gfx1250) — compile-verified
//
#include <hip/hip_runtime.h>
#include <hip/hip_bf16.h>
#include <stdint.h>

// ---- problem constants (from reference) ----
#define B_    16
#define N_    4096
#define S_    1024          // NPOINT
#define K_    32            // NSAMPLE
#define C_    64            // feature channels
#define C0_   67            // 3 + C
#define C0P_  68            // padded to multiple of 4 for WMMA K-steps
#define NTOT  524288L       // B_*S_*K_
#define RAD2  0.04f         // RADIUS^2
#define BNEPS 1e-5f

typedef __attribute__((ext_vector_type(2))) float v2f;
typedef __attribute__((ext_vector_type(8))) float v8f;

// ---------------------------------------------------------------------------
// 1) Farthest point sampling: one block per batch, xyz cached in LDS.
//    Reference scan emits the *previous* "last" each step (idx[0]==0).
// ---------------------------------------------------------------------------
__global__ __launch_bounds__(512) void fps_kernel(const float* __restrict__ xyz,
                                                  int* __restrict__ fps_idx,
                                                  float* __restrict__ new_xyz) {
  __shared__ float sx[N_ * 3];        // 48 KB of 320 KB WGP LDS
  __shared__ float sval[512];
  __shared__ int   sidx[512];
  const int b   = blockIdx.x;
  const int tid = threadIdx.x;
  const float* src = xyz + (size_t)b * N_ * 3;
  for (int i = tid; i < N_ * 3; i += 512) sx[i] = src[i];
  __syncthreads();

  float dist[N_ / 512];
#pragma unroll
  for (int t = 0; t < N_ / 512; ++t) dist[t] = 1e10f;

  int last = 0;
  for (int step = 0; step < S_; ++step) {
    if (tid == 0) {
      fps_idx[b * S_ + step] = last;
      new_xyz[((size_t)b * S_ + step) * 3 + 0] = sx[last * 3 + 0];
      new_xyz[((size_t)b * S_ + step) * 3 + 1] = sx[last * 3 + 1];
      new_xyz[((size_t)b * S_ + step) * 3 + 2] = sx[last * 3 + 2];
    }
    const float cx = sx[last * 3 + 0], cy = sx[last * 3 + 1], cz = sx[last * 3 + 2];
    float bv = -1.0f;
    int   bi = 0x7fffffff;
#pragma unroll
    for (int t = 0; t < N_ / 512; ++t) {
      const int j = t * 512 + tid;
      const float dx = sx[j * 3 + 0] - cx;
      const float dy = sx[j * 3 + 1] - cy;
      const float dz = sx[j * 3 + 2] - cz;
      const float nd = fminf(dist[t], dx * dx + dy * dy + dz * dz);
      dist[t] = nd;
      if (nd > bv) { bv = nd; bi = j; }   // ascending j -> first-max within thread
    }
    sval[tid] = bv; sidx[tid] = bi;
    __syncthreads();
    for (int o = 256; o > 0; o >>= 1) {   // argmax, ties -> smaller index (jnp.argmax)
      if (tid < o) {
        const float v = sval[tid + o]; const int ii = sidx[tid + o];
        if (v > sval[tid] || (v == sval[tid] && ii < sidx[tid])) { sval[tid] = v; sidx[tid] = ii; }
      }
      __syncthreads();
    }
    last = sidx[0];
    __syncthreads();
  }
}

// ---------------------------------------------------------------------------
// 2) Ball query: one wave32 per (b,s); ballot + prefix-popcount emits the
//    first 32 in-radius indices in ascending order (== reference sort+take).
// ---------------------------------------------------------------------------
__global__ __launch_bounds__(256) void ball_query_kernel(const float* __restrict__ xyz,
                                                         const float* __restrict__ new_xyz,
                                                         int* __restrict__ idx) {
  const int gw   = (blockIdx.x * blockDim.x + threadIdx.x) >> 5;  // wave-uniform
  const int lane = threadIdx.x & 31;
  if (gw >= B_ * S_) return;
  const int b = gw / S_, s = gw % S_;
  const float cx = new_xyz[((size_t)b * S_ + s) * 3 + 0];
  const float cy = new_xyz[((size_t)b * S_ + s) * 3 + 1];
  const float cz = new_xyz[((size_t)b * S_ + s) * 3 + 2];
  const float* base = xyz + (size_t)b * N_ * 3;
  int* out = idx + ((size_t)b * S_ + s) * K_;

  int cnt = 0, first = 0, havefirst = 0;
  for (int j0 = 0; j0 < N_ && cnt < K_; j0 += 32) {
    const int j = j0 + lane;
    const float dx = base[j * 3 + 0] - cx;
    const float dy = base[j * 3 + 1] - cy;
    const float dz = base[j * 3 + 2] - cz;
    const bool in = (dx * dx + dy * dy + dz * dz) < RAD2;
    const unsigned m = __builtin_amdgcn_ballot_w32(in);
    if (m) {
      if (!havefirst) { first = j0 + __builtin_ctz(m); havefirst = 1; }
      const unsigned below = m & ((1u << lane) - 1u);
      if (in) {
        const int slot = cnt + __popc(below);
        if (slot < K_) out[slot] = j;
      }
      cnt += __popc(m);
    }
  }
  const int pad   = havefirst ? first : 0;
  const int start = cnt < K_ ? cnt : K_;
  for (int t = start + lane; t < K_; t += 32) out[t] = pad;
}

// ---------------------------------------------------------------------------
// 3) Gather + center + concat -> x0 [C0P_][NTOT] row-major (coalesced over n)
// ---------------------------------------------------------------------------
__global__ __launch_bounds__(256) void build_x0_kernel(const float* __restrict__ xyz,
                                                       const float* __restrict__ feat,
                                                       const float* __restrict__ new_xyz,
                                                       const int* __restrict__ idx,
                                                       float* __restrict__ x0) {
  const long n = (long)blockIdx.x * 256 + threadIdx.x;
  if (n >= NTOT) return;
  const long bs = n >> 5;                 // n / K_
  const int  s  = (int)(bs % S_);
  const int  b  = (int)(bs / S_);
  const int  j  = idx[n];
  const float* p = xyz + ((size_t)b * N_ + j) * 3;
  const float* q = new_xyz + ((size_t)b * S_ + s) * 3;
  x0[0 * NTOT + n] = p[0] - q[0];
  x0[1 * NTOT + n] = p[1] - q[1];
  x0[2 * NTOT + n] = p[2] - q[2];
  const float* f = feat + (size_t)b * C_ * N_ + j;
#pragma unroll 8
  for (int c = 0; c < C_; ++c) x0[(size_t)(3 + c) * NTOT + n] = f[(size_t)c * N_];
  for (int c = 3 + C_; c < C0P_; ++c) x0[(size_t)c * NTOT + n] = 0.f;
}

// ---------------------------------------------------------------------------
// Weight pad/copy (cin -> cinP, zero padded K columns)
// ---------------------------------------------------------------------------
__global__ void wpad_kernel(const float* __restrict__ w, float* __restrict__ wp,
                            int cout, int cin, int cinP) {
  const int i = blockIdx.x * blockDim.x + threadIdx.x;
  if (i >= cout * cinP) return;
  const int o = i / cinP, c = i % cinP;
  wp[i] = (c < cin) ? w[o * cin + c] : 0.f;
}

// ---------------------------------------------------------------------------
// 4) WMMA GEMM, column-strip form: each wave owns ONE 16-column group of X
//    and ALL cout/16 row-tiles (OTILES accumulators in registers).
//    Per K-step: 1 B-fragment load reused across OTILES WMMAs (A comes from
//    the L1-resident weight matrix). Cuts X L2 traffic by OTILES x and gives
//    the WMMA pipeline an OTILES-deep dependency distance.
//    Fragment layouts per CDNA5 ISA 7.12.2 (f32 A 16x4, B 4x16, C/D 16x16).
//    Early-exit is wave-uniform so EXEC is all-ones for every WMMA.
// ---------------------------------------------------------------------------
template <int OTILES, int CINP>
__global__ __launch_bounds__(256) void gemm_wmma_strip_kernel(const float* __restrict__ W,
                                                              const float* __restrict__ X,
                                                              const float* __restrict__ bias,
                                                              float* __restrict__ Y) {
  const int  lane = threadIdx.x & 31;
  const long wave = (long)blockIdx.x * 8 + (threadIdx.x >> 5);
  const long ntiles = NTOT >> 4;
  if (wave >= ntiles) return;           // wave-uniform

  const int  m    = lane & 15;          // A row / B,C,D column within tile
  const int  kh   = (lane >> 4) << 1;   // lanes 16-31 hold K+2,K+3
  const long ncol = wave * 16 + m;
  const float* xcol = X + ncol;

  v8f acc[OTILES];
#pragma unroll
  for (int t = 0; t < OTILES; ++t) acc[t] = (v8f){0.f,0.f,0.f,0.f,0.f,0.f,0.f,0.f};

#pragma unroll
  for (int k0 = 0; k0 < CINP; k0 += 4) {
    v2f bm;
    bm.x = xcol[(size_t)(k0 + kh) * NTOT];
    bm.y = xcol[(size_t)(k0 + kh + 1) * NTOT];
#pragma unroll
    for (int t = 0; t < OTILES; ++t) {
      const float* wrow = W + (size_t)(t * 16 + m) * CINP;
      v2f a;
      a.x = wrow[k0 + kh];
      a.y = wrow[k0 + kh + 1];
      acc[t] = __builtin_amdgcn_wmma_f32_16x16x4_f32(
          /*neg_a=*/false, a, /*neg_b=*/false, bm,
          /*c_mod=*/(short)0, acc[t], /*reuse_a=*/false, /*reuse_b=*/false);
    }
  }

  const int hi8 = (lane >> 4) << 3;     // lanes 16-31 hold rows M+8
#pragma unroll
  for (int t = 0; t < OTILES; ++t) {
#pragma unroll
    for (int r = 0; r < 8; ++r) {
      const int orow = t * 16 + r + hi8;
      Y[(size_t)orow * NTOT + ncol] = acc[t][r] + bias[orow];
    }
  }
}

// ---------------------------------------------------------------------------
// Batch-norm statistics: sum & sumsq per channel (population var = E[x^2]-m^2)
// ---------------------------------------------------------------------------
__global__ void zero_stats_kernel(float* __restrict__ stats, int n) {
  for (int i = threadIdx.x; i < n; i += blockDim.x) stats[i] = 0.f;
}

__global__ __launch_bounds__(256) void bn_reduce_kernel(const float* __restrict__ Y,
                                                        float* __restrict__ stats,
                                                        int cout) {
  __shared__ float ss[256], sq[256];
  const int  chunks = (int)(NTOT / 8192);
  const int  c  = blockIdx.x / chunks;
  const long n0 = (long)(blockIdx.x % chunks) * 8192;
  const float* row = Y + (size_t)c * NTOT + n0;
  float s = 0.f, q = 0.f;
  for (int t = threadIdx.x; t < 8192; t += 256) {
    const float v = row[t];
    s += v; q += v * v;
  }
  ss[threadIdx.x] = s; sq[threadIdx.x] = q;
  __syncthreads();
  for (int o = 128; o > 0; o >>= 1) {
    if ((int)threadIdx.x < o) {
      ss[threadIdx.x] += ss[threadIdx.x + o];
      sq[threadIdx.x] += sq[threadIdx.x + o];
    }
    __syncthreads();
  }
  if (threadIdx.x == 0) {
    atomicAdd(&stats[c], ss[0]);
    atomicAdd(&stats[cout + c], sq[0]);
  }
}

// BN + ReLU -> next layer input (layers 0,1)
__global__ __launch_bounds__(256) void bn_relu_apply_kernel(const float* __restrict__ Y,
                                                            const float* __restrict__ stats,
                                                            const float* __restrict__ g,
                                                            const float* __restrict__ be,
                                                            float* __restrict__ xout,
                                                            int cout) {
  const long i = (long)blockIdx.x * 256 + threadIdx.x;
  if (i >= (long)cout * NTOT) return;
  const int c = (int)(i / NTOT);
  const float inv_n = 1.0f / (float)NTOT;
  const float mean  = stats[c] * inv_n;
  const float var   = stats[cout + c] * inv_n - mean * mean;
  const float rs    = rsqrtf(var + BNEPS);
  const float r = g[c] * ((Y[i] - mean) * rs) + be[c];
  xout[i] = fmaxf(r, 0.f);
}

// Final layer: BN + ReLU fused with max over K_ -> new_features (B, cout, S)
__global__ __launch_bounds__(256) void bn_relu_max_kernel(const float* __restrict__ Y,
                                                          const float* __restrict__ stats,
                                                          const float* __restrict__ g,
                                                          const float* __restrict__ be,
                                                          float* __restrict__ out,
                                                          int cout) {
  const long i = (long)blockIdx.x * 256 + threadIdx.x;
  const long total = (long)B_ * cout * S_;
  if (i >= total) return;
  const int  s = (int)(i % S_);
  const long t = i / S_;
  const int  c = (int)(t % cout);
  const int  b = (int)(t / cout);
  const float inv_n = 1.0f / (float)NTOT;
  const float mean  = stats[c] * inv_n;
  const float var   = stats[cout + c] * inv_n - mean * mean;
  const float rs    = rsqrtf(var + BNEPS);
  const float gg = g[c], bb = be[c];
  const float* row = Y + (size_t)c * NTOT + ((size_t)b * S_ + s) * K_;
  float mx = 0.f;                        // ReLU output is >= 0
#pragma unroll
  for (int k = 0; k < K_; ++k) {
    const float r = fmaxf(gg * ((row[k] - mean) * rs) + bb, 0.f);
    mx = fmaxf(mx, r);
  }
  out[i] = mx;                           // layout ((b*cout)+c)*S + s == i
}

// ---------------------------------------------------------------------------
// Host launcher
// ---------------------------------------------------------------------------
extern "C" void kernel_launch(void* const* d_in, const int* in_sizes, int n_in,
                              void* d_out, int out_size, void* d_ws, size_t ws_size,
                              hipStream_t stream) {
  const float* xyz  = (const float*)d_in[0];
  const float* feat = (const float*)d_in[1];
  const float* w[3]  = {(const float*)d_in[2], (const float*)d_in[6],  (const float*)d_in[10]};
  const float* bs[3] = {(const float*)d_in[3], (const float*)d_in[7],  (const float*)d_in[11]};
  const float* g[3]  = {(const float*)d_in[4], (const float*)d_in[8],  (const float*)d_in[12]};
  const float* be[3] = {(const float*)d_in[5], (const float*)d_in[9],  (const float*)d_in[13]};

  char* ws = (char*)d_ws;
  size_t off = 0;
  auto take = [&](size_t bytes) -> char* {
    char* p = ws + off;
    off = (off + bytes + 255) & ~(size_t)255;
    return p;
  };
  int*   fps_idx  = (int*)take((size_t)B_ * S_ * 4);
  int*   ball_idx = (int*)take((size_t)B_ * S_ * K_ * 4);
  float* stats    = (float*)take(2 * 128 * 4);
  float* wp0      = (float*)take((size_t)64 * C0P_ * 4);
  float* wp1      = (float*)take((size_t)64 * 64 * 4);
  float* wp2      = (float*)take((size_t)128 * 64 * 4);
  float* xact     = (float*)take((size_t)C0P_ * NTOT * 4);   // ~142.6 MB
  float* ybuf     = (float*)take((size_t)128 * NTOT * 4);    // ~268.4 MB
  (void)ws_size; (void)in_sizes; (void)n_in; (void)out_size;

  float* new_xyz  = (float*)d_out;                            // (B, S, 3)
  float* new_feat = (float*)d_out + (size_t)B_ * S_ * 3;      // (B, 128, S)

  fps_kernel<<<B_, 512, 0, stream>>>(xyz, fps_idx, new_xyz);

  {
    const int threads = B_ * S_ * 32;                         // one wave per (b,s)
    ball_query_kernel<<<threads / 256, 256, 0, stream>>>(xyz, new_xyz, ball_idx);
  }

  build_x0_kernel<<<(int)(NTOT / 256), 256, 0, stream>>>(xyz, feat, new_xyz, ball_idx, xact);

  wpad_kernel<<<(64 * C0P_ + 255) / 256, 256, 0, stream>>>(w[0], wp0, 64, C0_, C0P_);
  wpad_kernel<<<(64 * 64 + 255) / 256, 256, 0, stream>>>(w[1], wp1, 64, 64, 64);
  wpad_kernel<<<(128 * 64 + 255) / 256, 256, 0, stream>>>(w[2], wp2, 128, 64, 64);

  const int    coutL[3] = {64, 64, 128};
  const int    gemm_blocks = (int)((NTOT / 16) / 8);          // 1 wave per 16-col strip
  const float* Xl = xact;

  for (int l = 0; l < 3; ++l) {
    if (l == 0) {
      gemm_wmma_strip_kernel<4, C0P_><<<gemm_blocks, 256, 0, stream>>>(wp0, Xl, bs[0], ybuf);
    } else if (l == 1) {
      gemm_wmma_strip_kernel<4, 64><<<gemm_blocks, 256, 0, stream>>>(wp1, Xl, bs[1], ybuf);
    } else {
      gemm_wmma_strip_kernel<8, 64><<<gemm_blocks, 256, 0, stream>>>(wp2, Xl, bs[2], ybuf);
    }

    zero_stats_kernel<<<1, 256, 0, stream>>>(stats, 2 * coutL[l]);
    bn_reduce_kernel<<<coutL[l] * (int)(NTOT / 8192), 256, 0, stream>>>(ybuf, stats, coutL[l]);

    if (l < 2) {
      const long total = (long)coutL[l] * NTOT;
      bn_relu_apply_kernel<<<(int)((total + 255) / 256), 256, 0, stream>>>(
          ybuf, stats, g[l], be[l], xact, coutL[l]);
      Xl = xact;
    } else {
      const long total = (long)B_ * coutL[l] * S_;
      bn_relu_max_kernel<<<(int)((total + 255) / 256), 256, 0, stream>>>(
          ybuf, stats, g[l], be[l], new_feat, coutL[l]);
    }
  }
}